// SuperGlue_81037442941186
// MI455X (gfx1250) — compile-verified
//
#include <hip/hip_runtime.h>
#include <math.h>

// ---------------------------------------------------------------------------
// SuperGlue forward for MI455X (gfx1250, wave32, WMMA)
//   D=26, H=2, DH=13, L=18, N=M=2048, SINKHORN=100
// ---------------------------------------------------------------------------

typedef __attribute__((ext_vector_type(2))) float v2f;
typedef __attribute__((ext_vector_type(8))) float v8f;

#define NPOS 2048
#define CH   26
#define NB   2049   // N+1 (Sinkhorn augmented size)

__device__ __forceinline__ v8f wmma4(v2f a, v2f b, v8f c) {
    // D = A(16x4 f32) * B(4x16 f32) + C(16x16 f32)
    return __builtin_amdgcn_wmma_f32_16x16x4_f32(
        false, a, false, b, (short)0, c, false, false);
}

__device__ __forceinline__ v8f v8f_zero() {
    v8f z;
#pragma unroll
    for (int r = 0; r < 8; ++r) z[r] = 0.f;
    return z;
}

// Branch-free masked load: always loads a valid (clamped) address, selects 0.
__device__ __forceinline__ float ldmask(const float* __restrict__ p, size_t idx_ok,
                                        bool valid) {
    float x = p[idx_ok];
    return valid ? x : 0.f;
}

// ---------------------------------------------------------------------------
// Generic 1x1 conv: y[c,n] = W[c,:] . concat(x0,x1)[:,n] + b[c], optional relu
// grid = (2048/256, Cout), block = 256
// ---------------------------------------------------------------------------
__global__ void conv1_kernel(const float* __restrict__ W, const float* __restrict__ bias,
                             const float* __restrict__ x0, const float* __restrict__ x1,
                             float* __restrict__ y, int Cout, int Cin0, int Cin1, int relu) {
    int n = blockIdx.x * 256 + threadIdx.x;
    int c = blockIdx.y;
    int Cin = Cin0 + Cin1;
    const float* wr = W + (size_t)c * Cin;
    float acc = bias[c];
    for (int i = 0; i < Cin0; ++i) acc += wr[i] * x0[(size_t)i * NPOS + n];
    for (int i = 0; i < Cin1; ++i) acc += wr[Cin0 + i] * x1[(size_t)i * NPOS + n];
    if (relu) acc = fmaxf(acc, 0.f);
    y[(size_t)c * NPOS + n] = acc;
}

__global__ void add_inplace_kernel(float* __restrict__ d, const float* __restrict__ delta) {
    int i = blockIdx.x * 256 + threadIdx.x;
    d[i] += delta[i];
}

__global__ void zero_kernel(float* __restrict__ p, int n) {
    int i = blockIdx.x * 256 + threadIdx.x;
    if (i < n) p[i] = 0.f;
}

// ---------------------------------------------------------------------------
// Keypoint encoder: MLP [3->6->13->26] with InstanceNorm+ReLU between layers,
// result added to desc.  One block (256 threads) per image; intermediates in
// global scratch, stats via LDS tree reduction.
// ---------------------------------------------------------------------------
__device__ void block_stats(const float* __restrict__ v, float* mean_out, float* rstd_out,
                            float* rsum, float* rsq) {
    float s = 0.f, q = 0.f;
    for (int n = threadIdx.x; n < NPOS; n += 256) { float x = v[n]; s += x; q += x * x; }
    rsum[threadIdx.x] = s; rsq[threadIdx.x] = q;
    __syncthreads();
    for (int k = 128; k > 0; k >>= 1) {
        if (threadIdx.x < (unsigned)k) {
            rsum[threadIdx.x] += rsum[threadIdx.x + k];
            rsq[threadIdx.x]  += rsq[threadIdx.x + k];
        }
        __syncthreads();
    }
    if (threadIdx.x == 0) {
        float mu  = rsum[0] * (1.f / NPOS);
        float var = rsq[0] * (1.f / NPOS) - mu * mu;
        *mean_out = mu;
        *rstd_out = rsqrtf(var + 1e-5f);
    }
    __syncthreads();
}

__global__ __launch_bounds__(256) void kenc_kernel(
    const float* __restrict__ kpts, const float* __restrict__ scores,
    const float* __restrict__ desc,
    const float* __restrict__ w0, const float* __restrict__ b0,
    const float* __restrict__ w1, const float* __restrict__ b1,
    const float* __restrict__ w2, const float* __restrict__ b2,
    float* __restrict__ y0, float* __restrict__ y1, float* __restrict__ dout) {
    __shared__ float rsum[256], rsq[256];
    __shared__ float smean[13], srstd[13];
    const int tid = threadIdx.x;

    // layer0: 3 -> 6
    for (int idx = tid; idx < 6 * NPOS; idx += 256) {
        int c = idx >> 11, n = idx & (NPOS - 1);
        float px = (kpts[n * 2 + 0] - 320.f) * (1.f / 448.f);
        float py = (kpts[n * 2 + 1] - 240.f) * (1.f / 448.f);
        float sc = scores[n];
        y0[idx] = w0[c * 3 + 0] * px + w0[c * 3 + 1] * py + w0[c * 3 + 2] * sc + b0[c];
    }
    __syncthreads();
    for (int c = 0; c < 6; ++c) block_stats(y0 + (size_t)c * NPOS, &smean[c], &srstd[c], rsum, rsq);

    // layer1: 6 -> 13 (inorm+relu on inputs)
    for (int idx = tid; idx < 13 * NPOS; idx += 256) {
        int c = idx / NPOS, n = idx % NPOS;
        float acc = b1[c];
        for (int i = 0; i < 6; ++i) {
            float x = fmaxf((y0[(size_t)i * NPOS + n] - smean[i]) * srstd[i], 0.f);
            acc += w1[c * 6 + i] * x;
        }
        y1[idx] = acc;
    }
    __syncthreads();
    for (int c = 0; c < 13; ++c) block_stats(y1 + (size_t)c * NPOS, &smean[c], &srstd[c], rsum, rsq);

    // layer2: 13 -> 26, add descriptor
    for (int idx = tid; idx < 26 * NPOS; idx += 256) {
        int c = idx / NPOS, n = idx % NPOS;
        float acc = b2[c];
        for (int i = 0; i < 13; ++i) {
            float x = fmaxf((y1[(size_t)i * NPOS + n] - smean[i]) * srstd[i], 0.f);
            acc += w2[c * 13 + i] * x;
        }
        dout[idx] = desc[idx] + acc;
    }
}

// ---------------------------------------------------------------------------
// Flash attention, f32 WMMA 16x16x4.
//   q,k,v: [26,2048] head-interleaved (channel = d*2+h), DH=13.
//   Computes S^T tiles (key=M rows, query=cols) so softmax-over-keys is an
//   in-register reduction + one shfl_xor(16).  Edge mask applied POST-softmax
//   (numerator only), matching the reference.
//   All fragment loads are branch-free (clamped address + cndmask zero).
// grid = (2048/(8*16), H), block = 256 (8 waves, one 16-query tile each)
// ---------------------------------------------------------------------------
__global__ __launch_bounds__(256) void attn_kernel(
    const float* __restrict__ qm, const float* __restrict__ km,
    const float* __restrict__ vm, const float* __restrict__ edge,
    float* __restrict__ out, int use_edge) {
    const int lane = threadIdx.x & 31;
    const int wave = threadIdx.x >> 5;
    const int col  = lane & 15;   // query column within tile (and M row for A frags)
    const int grp  = lane >> 4;   // half-wave group
    const int h    = blockIdx.y;
    const int qbase = (blockIdx.x * 8 + wave) * 16;
    const float scale = 0.2773500981126146f;  // 1/sqrt(13)

    // Per-fragment d indices (k-dim of WMMA): d = ks*4 + j + 2*grp, clamped.
    int dA[4], dB[4];       // channel rows for .x / .y of each k-step
    bool vA[4], vB[4];      // validity (d < 13)
#pragma unroll
    for (int ks = 0; ks < 4; ++ks) {
        int d0 = ks * 4 + 2 * grp;
        vA[ks] = d0 < 13;      dA[ks] = (vA[ks] ? d0 : 12) * 2 + h;
        vB[ks] = (d0 + 1) < 13; dB[ks] = (vB[ks] ? d0 + 1 : 12) * 2 + h;
    }
    // V-tile A-fragment row: M index = d = col, clamped.
    const bool vcol = col < 13;
    const size_t vrow = (size_t)((vcol ? col : 12) * 2 + h) * NPOS;

    // B fragments of Q: B[k=d][col=query]
    v2f bq[4];
#pragma unroll
    for (int ks = 0; ks < 4; ++ks) {
        v2f t;
        t.x = ldmask(qm, (size_t)dA[ks] * NPOS + qbase + col, vA[ks]);
        t.y = ldmask(qm, (size_t)dB[ks] * NPOS + qbase + col, vB[ks]);
        bq[ks] = t;
    }

    float run_max = -__builtin_inff();
    float run_sum = 0.f;
    v8f   o_acc   = v8f_zero();

    for (int mb = 0; mb < NPOS; mb += 16) {
        if (mb + 16 < NPOS) __builtin_prefetch(km + (size_t)h * NPOS + mb + 16 + col, 0, 1);

        // --- S^T tile = K_tile^T x Q  (A: M=key, K=d) ---
        v8f s = v8f_zero();
#pragma unroll
        for (int ks = 0; ks < 4; ++ks) {
            v2f ak;
            ak.x = ldmask(km, (size_t)dA[ks] * NPOS + mb + col, vA[ks]);
            ak.y = ldmask(km, (size_t)dB[ks] * NPOS + mb + col, vB[ks]);
            s = wmma4(ak, bq[ks], s);
        }

        // --- online softmax over keys (rows of S^T) ---
        float p[8];
        float tmax = -__builtin_inff();
#pragma unroll
        for (int r = 0; r < 8; ++r) { p[r] = s[r] * scale; tmax = fmaxf(tmax, p[r]); }
        tmax = fmaxf(tmax, __shfl_xor(tmax, 16, 32));
        float nmax = fmaxf(run_max, tmax);
        float f = expf(run_max - nmax);
        float ls = 0.f;
#pragma unroll
        for (int r = 0; r < 8; ++r) { p[r] = expf(p[r] - nmax); ls += p[r]; }
        ls += __shfl_xor(ls, 16, 32);
        run_sum = run_sum * f + ls;
        run_max = nmax;

        // --- post-softmax edge mask (numerator only) ---
        if (use_edge) {
            const float4* ep = (const float4*)(edge + (size_t)(qbase + col) * NPOS + mb + 8 * grp);
            float4 e0 = ep[0], e1 = ep[1];
            float ev[8] = {e0.x, e0.y, e0.z, e0.w, e1.x, e1.y, e1.z, e1.w};
#pragma unroll
            for (int r = 0; r < 8; ++r) p[r] *= (ev[r] == 1.0f) ? 0.7f : ev[r];
        }

#pragma unroll
        for (int r = 0; r < 8; ++r) o_acc[r] *= f;

        // --- reshape P^T D-tile into B fragments (K = key dim) ---
        float sw[8];
#pragma unroll
        for (int r = 0; r < 8; ++r) sw[r] = __shfl_xor(p[r], 16, 32);
        v2f bp[4];
        bp[0].x = grp ? sw[2] : p[0];  bp[0].y = grp ? sw[3] : p[1];
        bp[1].x = grp ? sw[6] : p[4];  bp[1].y = grp ? sw[7] : p[5];
        bp[2].x = grp ? p[2]  : sw[0]; bp[2].y = grp ? p[3]  : sw[1];
        bp[3].x = grp ? p[6]  : sw[4]; bp[3].y = grp ? p[7]  : sw[5];

        // --- O^T += V_tile x P^T  (A: M=d, K=key) ---
#pragma unroll
        for (int ks = 0; ks < 4; ++ks) {
            int m0 = mb + ks * 4 + 2 * grp;
            v2f av;
            av.x = ldmask(vm, vrow + m0,     vcol);
            av.y = ldmask(vm, vrow + m0 + 1, vcol);
            o_acc = wmma4(av, bp[ks], o_acc);
        }
    }

    float inv = 1.0f / run_sum;
#pragma unroll
    for (int r = 0; r < 8; ++r) {
        int d = r + 8 * grp;
        if (d < 13) out[(size_t)(d * 2 + h) * NPOS + qbase + col] = o_acc[r] * inv;
    }
}

// ---------------------------------------------------------------------------
// Score GEMM via WMMA:  Z[n,m] = md0[:,n].md1[:,m] / sqrt(26) into the
// augmented [2049 x 2049] Sinkhorn matrix (top-left 2048x2048 block).
// grid = 2048, block = 256 (8 waves, one 16x16 tile each)
// ---------------------------------------------------------------------------
__global__ __launch_bounds__(256) void score_kernel(
    const float* __restrict__ a_mat, const float* __restrict__ b_mat,
    float* __restrict__ Z) {
    const int lane = threadIdx.x & 31;
    const int wave = threadIdx.x >> 5;
    const int col  = lane & 15;
    const int grp  = lane >> 4;
    const int tile = blockIdx.x * 8 + wave;
    const int nb   = (tile >> 7) * 16;
    const int mb   = (tile & 127) * 16;
    const float scale = 0.19611613513818404f;  // 1/sqrt(26)

    v8f acc = v8f_zero();
#pragma unroll
    for (int ks = 0; ks < 7; ++ks) {  // K = 26 padded to 28
        int d0 = ks * 4 + 2 * grp;
        bool va = d0 < 26, vb = (d0 + 1) < 26;
        size_t ra = (size_t)(va ? d0 : 25) * NPOS;
        size_t rb = (size_t)(vb ? d0 + 1 : 25) * NPOS;
        v2f aa, bb;
        aa.x = ldmask(a_mat, ra + nb + col, va);
        aa.y = ldmask(a_mat, rb + nb + col, vb);
        bb.x = ldmask(b_mat, ra + mb + col, va);
        bb.y = ldmask(b_mat, rb + mb + col, vb);
        acc = wmma4(aa, bb, acc);
    }
#pragma unroll
    for (int r = 0; r < 8; ++r) {
        int n = nb + r + 8 * grp;
        Z[(size_t)n * NB + mb + col] = acc[r] * scale;
    }
}

__global__ void fill_bins_kernel(float* __restrict__ Z, const float* __restrict__ alpha_p) {
    float a = *alpha_p;
    int t = blockIdx.x * 256 + threadIdx.x;
    if (t < NB) {
        Z[(size_t)NPOS * NB + t] = a;                       // last row (incl. corner)
    } else if (t < NB + NPOS) {
        int i = t - NB;
        Z[(size_t)i * NB + NPOS] = a;                       // last column
    }
}

// ---------------------------------------------------------------------------
// Sinkhorn in log space.  norm = -log(m+n) = -log(4096); bin mu/nu = -log(2).
// ---------------------------------------------------------------------------
#define K_NORM     (-8.3177661667193436f)
#define K_BIN      (-0.6931471805599453f)

__global__ __launch_bounds__(256) void row_lse_kernel(
    const float* __restrict__ Z, const float* __restrict__ vvec, float* __restrict__ uvec) {
    __shared__ float sm[256], ss[256];
    const int i = blockIdx.x;
    const float* row = Z + (size_t)i * NB;
    float m = -__builtin_inff(), s = 0.f;
    for (int j = threadIdx.x; j < NB; j += 256) {
        float z = row[j] + vvec[j];
        if (z > m) { s = s * expf(m - z) + 1.f; m = z; }
        else       { s += expf(z - m); }
    }
    sm[threadIdx.x] = m; ss[threadIdx.x] = s;
    __syncthreads();
    for (int k = 128; k > 0; k >>= 1) {
        if (threadIdx.x < (unsigned)k) {
            float m1 = sm[threadIdx.x], s1 = ss[threadIdx.x];
            float m2 = sm[threadIdx.x + k], s2 = ss[threadIdx.x + k];
            float M = fmaxf(m1, m2);
            sm[threadIdx.x] = M;
            ss[threadIdx.x] = s1 * expf(m1 - M) + s2 * expf(m2 - M);
        }
        __syncthreads();
    }
    if (threadIdx.x == 0) {
        float lmu = (i < NPOS) ? K_NORM : K_BIN;
        uvec[i] = lmu - (sm[0] + logf(ss[0]));
    }
}

__global__ __launch_bounds__(256) void col_lse_kernel(
    const float* __restrict__ Z, const float* __restrict__ uvec, float* __restrict__ vvec) {
    int j = blockIdx.x * 256 + threadIdx.x;
    if (j >= NB) return;
    float m = -__builtin_inff(), s = 0.f;
    for (int i = 0; i < NB; ++i) {
        float z = Z[(size_t)i * NB + j] + uvec[i];
        if (z > m) { s = s * expf(m - z) + 1.f; m = z; }
        else       { s += expf(z - m); }
    }
    float lnu = (j < NPOS) ? K_NORM : K_BIN;
    vvec[j] = lnu - (m + logf(s));
}

__global__ void final_out_kernel(const float* __restrict__ Z, const float* __restrict__ uvec,
                                 const float* __restrict__ vvec, float* __restrict__ out) {
    size_t idx = (size_t)blockIdx.x * 256 + threadIdx.x;
    if (idx < (size_t)NB * NB) {
        int i = (int)(idx / NB), j = (int)(idx % NB);
        out[idx] = Z[idx] + uvec[i] + vvec[j] + 8.3177661667193436f;  // - norm
    }
}

// ---------------------------------------------------------------------------
// Host driver
// ---------------------------------------------------------------------------
extern "C" void kernel_launch(void* const* d_in, const int* in_sizes, int n_in,
                              void* d_out, int out_size, void* d_ws, size_t ws_size,
                              hipStream_t stream) {
    (void)in_sizes; (void)n_in; (void)out_size; (void)ws_size;

    const float* desc0   = (const float*)d_in[0];
    const float* desc1   = (const float*)d_in[1];
    const float* kpts0   = (const float*)d_in[2];
    const float* kpts1   = (const float*)d_in[3];
    const float* scores0 = (const float*)d_in[4];
    const float* scores1 = (const float*)d_in[5];
    const float* edge0   = (const float*)d_in[6];
    const float* edge1   = (const float*)d_in[7];
    const float* kenc_w0 = (const float*)d_in[8];
    const float* kenc_b0 = (const float*)d_in[9];
    const float* kenc_w1 = (const float*)d_in[10];
    const float* kenc_b1 = (const float*)d_in[11];
    const float* kenc_w2 = (const float*)d_in[12];
    const float* kenc_b2 = (const float*)d_in[13];
    const float* proj_w  = (const float*)d_in[14];
    const float* proj_b  = (const float*)d_in[15];
    const float* merge_w = (const float*)d_in[16];
    const float* merge_b = (const float*)d_in[17];
    const float* mlp_w0  = (const float*)d_in[18];
    const float* mlp_b0  = (const float*)d_in[19];
    const float* mlp_w1  = (const float*)d_in[20];
    const float* mlp_b1  = (const float*)d_in[21];
    const float* final_w = (const float*)d_in[22];
    const float* final_b = (const float*)d_in[23];
    const float* bin_sc  = (const float*)d_in[24];

    const size_t MAT = (size_t)CH * NPOS;  // 53248
    float* W = (float*)d_ws;
    size_t off = 0;
    auto alloc = [&](size_t n) { float* p = W + off; off += n; return p; };

    float* d0  = alloc(MAT);     float* d1  = alloc(MAT);
    float* q0  = alloc(MAT);     float* k0  = alloc(MAT);   float* v0 = alloc(MAT);
    float* q1  = alloc(MAT);     float* k1  = alloc(MAT);   float* v1 = alloc(MAT);
    float* o0  = alloc(MAT);     float* o1  = alloc(MAT);
    float* m0  = alloc(MAT);     float* m1  = alloc(MAT);
    float* h0  = alloc(2 * MAT); float* h1  = alloc(2 * MAT);
    float* dl0 = alloc(MAT);     float* dl1 = alloc(MAT);
    float* y0a = alloc(6 * NPOS);  float* y1a = alloc(13 * NPOS);
    float* y0b = alloc(6 * NPOS);  float* y1b = alloc(13 * NPOS);
    float* md0 = alloc(MAT);     float* md1 = alloc(MAT);
    float* Z   = alloc((size_t)NB * NB);
    float* uv  = alloc(NB);      float* vv  = alloc(NB);

    const dim3 blk(256);

    // ---- keypoint encoder + descriptor add ----
    kenc_kernel<<<1, blk, 0, stream>>>(kpts0, scores0, desc0, kenc_w0, kenc_b0,
                                       kenc_w1, kenc_b1, kenc_w2, kenc_b2, y0a, y1a, d0);
    kenc_kernel<<<1, blk, 0, stream>>>(kpts1, scores1, desc1, kenc_w0, kenc_b0,
                                       kenc_w1, kenc_b1, kenc_w2, kenc_b2, y0b, y1b, d1);

    // ---- 18 GNN layers ----
    for (int l = 0; l < 18; ++l) {
        const int self = (l % 2 == 0);
        const float* pw = proj_w + (size_t)l * 3 * 26 * 26;
        const float* pb = proj_b + (size_t)l * 3 * 26;
        const float* src0 = self ? d0 : d1;
        const float* src1 = self ? d1 : d0;

        dim3 g26(8, 26), g52(8, 52);
        conv1_kernel<<<g26, blk, 0, stream>>>(pw,        pb,      d0,   nullptr, q0, 26, 26, 0, 0);
        conv1_kernel<<<g26, blk, 0, stream>>>(pw + 676,  pb + 26, src0, nullptr, k0, 26, 26, 0, 0);
        conv1_kernel<<<g26, blk, 0, stream>>>(pw + 1352, pb + 52, src0, nullptr, v0, 26, 26, 0, 0);
        conv1_kernel<<<g26, blk, 0, stream>>>(pw,        pb,      d1,   nullptr, q1, 26, 26, 0, 0);
        conv1_kernel<<<g26, blk, 0, stream>>>(pw + 676,  pb + 26, src1, nullptr, k1, 26, 26, 0, 0);
        conv1_kernel<<<g26, blk, 0, stream>>>(pw + 1352, pb + 52, src1, nullptr, v1, 26, 26, 0, 0);

        attn_kernel<<<dim3(16, 2), blk, 0, stream>>>(q0, k0, v0, edge0, o0, self);
        attn_kernel<<<dim3(16, 2), blk, 0, stream>>>(q1, k1, v1, edge1, o1, self);

        conv1_kernel<<<g26, blk, 0, stream>>>(merge_w + (size_t)l * 676, merge_b + (size_t)l * 26,
                                              o0, nullptr, m0, 26, 26, 0, 0);
        conv1_kernel<<<g26, blk, 0, stream>>>(merge_w + (size_t)l * 676, merge_b + (size_t)l * 26,
                                              o1, nullptr, m1, 26, 26, 0, 0);

        conv1_kernel<<<g52, blk, 0, stream>>>(mlp_w0 + (size_t)l * 2704, mlp_b0 + (size_t)l * 52,
                                              d0, m0, h0, 52, 26, 26, 1);
        conv1_kernel<<<g52, blk, 0, stream>>>(mlp_w0 + (size_t)l * 2704, mlp_b0 + (size_t)l * 52,
                                              d1, m1, h1, 52, 26, 26, 1);

        conv1_kernel<<<g26, blk, 0, stream>>>(mlp_w1 + (size_t)l * 1352, mlp_b1 + (size_t)l * 26,
                                              h0, nullptr, dl0, 26, 52, 0, 0);
        conv1_kernel<<<g26, blk, 0, stream>>>(mlp_w1 + (size_t)l * 1352, mlp_b1 + (size_t)l * 26,
                                              h1, nullptr, dl1, 26, 52, 0, 0);

        add_inplace_kernel<<<208, blk, 0, stream>>>(d0, dl0);
        add_inplace_kernel<<<208, blk, 0, stream>>>(d1, dl1);
    }

    // ---- final projection + score matrix ----
    dim3 g26(8, 26);
    conv1_kernel<<<g26, blk, 0, stream>>>(final_w, final_b, d0, nullptr, md0, 26, 26, 0, 0);
    conv1_kernel<<<g26, blk, 0, stream>>>(final_w, final_b, d1, nullptr, md1, 26, 26, 0, 0);
    score_kernel<<<2048, blk, 0, stream>>>(md0, md1, Z);
    fill_bins_kernel<<<(NB + NPOS + 255) / 256, blk, 0, stream>>>(Z, bin_sc);

    // ---- Sinkhorn (100 iterations) ----
    zero_kernel<<<(NB + 255) / 256, blk, 0, stream>>>(vv, NB);
    for (int it = 0; it < 100; ++it) {
        row_lse_kernel<<<NB, blk, 0, stream>>>(Z, vv, uv);
        col_lse_kernel<<<(NB + 255) / 256, blk, 0, stream>>>(Z, uv, vv);
    }
    final_out_kernel<<<(int)(((size_t)NB * NB + 255) / 256), blk, 0, stream>>>(
        Z, uv, vv, (float*)d_out);
}